// BinaryPositionEmbedding_11562051961176
// MI455X (gfx1250) — compile-verified
//
#include <hip/hip_runtime.h>

// BinaryPositionEmbedding as a skinny-K fp32 GEMM on the CDNA5 matrix pipe:
//   out[32768, 1024] = bits[32768, 13(->16)] @ emb[13(->16), 1024]
// using V_WMMA_F32_16X16X4_F32 (exact fp32, K padded 13->16; pad is free since
// x < 2^13 so bits 13..15 are always zero).
//
// Memory-bound: 128 MB of output stores -> ~5.5 us floor at 23.3 TB/s.
// B (embedding strip) lives in 8 VGPRs per wave for the whole kernel; A is
// synthesized from x bits in VALU; stores coalesce as 64B half-wave segments.

typedef __attribute__((ext_vector_type(2))) float v2f;
typedef __attribute__((ext_vector_type(8))) float v8f;

#define N_BITS     13
#define EMB_ROWS   (N_BITS + 1)     // table has 14 rows; rows 0..12 used
#define D_MODEL    1024
#define N_ROWS     32768            // 4 * 8192 flattened positions
#define ROW_BLOCKS (N_ROWS / 16)    // 2048
#define COL_TILES  (D_MODEL / 16)   // 64
#define WAVES_PER_BLOCK 8           // 256 threads, wave32

__global__ __launch_bounds__(256) void bpe_wmma_f32_kernel(
    const int*   __restrict__ x,    // [N_ROWS] position ids, < 2^13
    const float* __restrict__ emb,  // [EMB_ROWS, D_MODEL]
    float*       __restrict__ out)  // [N_ROWS, D_MODEL]
{
    const int lane  = threadIdx.x & 31;
    const int wave  = threadIdx.x >> 5;
    const int n     = lane & 15;   // B/C column within tile; A row within tile
    const int khalf = lane >> 4;   // half-wave selector (K split for A/B, M+8 for C)

    const int colBase = blockIdx.x * 16;

    // ---- B: 16x16 embedding strip (4 K-steps x v2f), resident all kernel ----
    // ISA layout (4x16 f32 B): lane L -> N = L&15, k = 2*(L>>4) + j.
    v2f b[4];
#pragma unroll
    for (int s = 0; s < 4; ++s) {
#pragma unroll
        for (int j = 0; j < 2; ++j) {
            const int kg  = s * 4 + khalf * 2 + j;              // global K: 0..15
            const int row = (kg < EMB_ROWS) ? kg : (EMB_ROWS - 1); // clamp; A==0 there
            b[s][j] = emb[row * D_MODEL + colBase + n];
        }
    }

    const int waveStride = gridDim.y * WAVES_PER_BLOCK;         // 512
    for (int rb = blockIdx.y * WAVES_PER_BLOCK + wave; rb < ROW_BLOCKS; rb += waveStride) {
        const int rowBase = rb * 16;

        // ---- A: bits of x, ISA layout (16x4 f32 A): lane L -> M = L&15,
        //         k = 2*(L>>4) + j. Both half-waves read the same 16 x values.
        const int xv = x[rowBase + n];
        v2f a[4];
#pragma unroll
        for (int s = 0; s < 4; ++s) {
#pragma unroll
            for (int j = 0; j < 2; ++j) {
                const int kg = s * 4 + khalf * 2 + j;
                a[s][j] = (float)((xv >> kg) & 1);   // kg>=13 -> always 0 (x < 2^13)
            }
        }

        // ---- 4 chained fp32 WMMAs: K = 16 ----
        v8f c = {};
#pragma unroll
        for (int s = 0; s < 4; ++s) {
            c = __builtin_amdgcn_wmma_f32_16x16x4_f32(
                    /*neg_a=*/false, a[s], /*neg_b=*/false, b[s],
                    /*c_mod=*/(short)0, c, /*reuse_a=*/false, /*reuse_b=*/false);
        }

        // ---- Store C: VGPR r -> row rowBase + r + 8*khalf, col colBase + n ----
        float* o = out + (size_t)(rowBase + khalf * 8) * D_MODEL + (colBase + n);
#pragma unroll
        for (int r = 0; r < 8; ++r) {
            o[(size_t)r * D_MODEL] = c[r];
        }
    }
}

extern "C" void kernel_launch(void* const* d_in, const int* in_sizes, int n_in,
                              void* d_out, int out_size, void* d_ws, size_t ws_size,
                              hipStream_t stream) {
    const int*   x   = (const int*)d_in[0];     // 32768 int32 position ids
    const float* emb = (const float*)d_in[1];   // 14 x 1024 fp32
    float*       out = (float*)d_out;           // 32768 x 1024 fp32

    dim3 grid(COL_TILES, 64);   // 64 col tiles x 64 row-slices = 4096 workgroups
    dim3 block(256);            // 8 waves (wave32)
    bpe_wmma_f32_kernel<<<grid, block, 0, stream>>>(x, emb, out);
}